// GNNBackBone_53858889892007
// MI455X (gfx1250) — compile-verified
//
#include <hip/hip_runtime.h>
#include <hip/hip_bf16.h>

// ---------------------------------------------------------------------------
// TAGConv x5 (K=3 hops) GNN backbone for MI455X (gfx1250, wave32).
//  - sparse propagation: float4 gather + native global_atomic_add_f32
//    (inline CDNA5 asm, device scope; L2-resident: 25.6MB features << 192MB L2)
//  - dense hop-GEMMs: fused over all 4 hops with V_WMMA_F32_16X16X4_F32
//    (full f32 precision matrix pipe), bias+ReLU in epilogue.
// ---------------------------------------------------------------------------

#define N_NODES 100000
#define E_EDGES 1600000
#define IN_DIM  32
#define LATENT  64

typedef __attribute__((ext_vector_type(2))) float v2f;
typedef __attribute__((ext_vector_type(8))) float v8f;

// Native FP32 atomic add (no return -> STOREcnt). Device scope: the L2 atomic
// units are the coherence point; results are consumed only by later kernels.
__device__ __forceinline__ void atomic_add_f32(float* p, float v) {
    asm volatile("global_atomic_add_f32 %0, %1, off scope:SCOPE_DEV"
                 :: "v"(p), "v"(v)
                 : "memory");
}

// ---------------- gcn_norm ----------------

__global__ void deg_kernel(const int* __restrict__ col,
                           const float* __restrict__ w,
                           float* __restrict__ deg) {
    int e = blockIdx.x * blockDim.x + threadIdx.x;
    if (e < E_EDGES) atomic_add_f32(&deg[col[e]], w[e]);
}

__global__ void dinv_kernel(float* __restrict__ deg) {
    int v = blockIdx.x * blockDim.x + threadIdx.x;
    if (v < N_NODES) {
        float d = deg[v];
        deg[v] = (d > 0.0f) ? __frsqrt_rn(d) : 0.0f;
    }
}

__global__ void norm_kernel(const int* __restrict__ row,
                            const int* __restrict__ col,
                            const float* __restrict__ w,
                            const float* __restrict__ dinv,
                            float* __restrict__ norm) {
    int e = blockIdx.x * blockDim.x + threadIdx.x;
    if (e < E_EDGES) norm[e] = dinv[row[e]] * w[e] * dinv[col[e]];
}

// ---------------- sparse propagation: dst[col] += norm * src[row] ----------
// 16 (or 8) consecutive lanes share one edge -> row/col/norm loads coalesce to
// a single request; payload is a 16B vector load + 4 native f32 atomics.

template <int D>
__global__ void scatter_kernel(const float* __restrict__ src,
                               float* __restrict__ dst,
                               const int* __restrict__ row,
                               const int* __restrict__ col,
                               const float* __restrict__ norm) {
    constexpr int VE = D / 4;              // float4 chunks per edge
    int t = blockIdx.x * blockDim.x + threadIdx.x;  // exact grid, no guard
    int e = t / VE;
    int q = t & (VE - 1);
    int r = row[e];
    int c = col[e];
    float nv = norm[e];
    const float4 v = *(const float4*)(src + (size_t)r * D + q * 4);
    float* dp = dst + (size_t)c * D + q * 4;
    atomic_add_f32(dp + 0, nv * v.x);
    atomic_add_f32(dp + 1, nv * v.y);
    atomic_add_f32(dp + 2, nv * v.z);
    atomic_add_f32(dp + 3, nv * v.w);
}

// ---------------- fused hop-GEMM: out = relu(sum_s H_s @ W_s + b) ----------
// One wave computes one 16x16 f32 tile via V_WMMA_F32_16X16X4_F32.
// Block = 256 threads = 8 waves = 2 row-tiles x 4 col-tiles (32 rows x 64 cols).

template <int D>
__global__ void __launch_bounds__(256)
tag_gemm_kernel(const float* __restrict__ H0, const float* __restrict__ H1,
                const float* __restrict__ H2, const float* __restrict__ H3,
                const float* __restrict__ W,      // (4, D, 64) row-major
                const float* __restrict__ bias,   // (64,)
                float* __restrict__ out) {        // (N, 64)
    const int wave  = threadIdx.x >> 5;
    const int lane  = threadIdx.x & 31;
    const int mtile = blockIdx.x * 2 + (wave >> 2);
    const int ntile = wave & 3;

    const int m    = lane & 15;       // A row within tile / B,C column index
    const int half = lane >> 4;       // lane half selects K pair / M+8
    const int arow = mtile * 16 + m;  // global A row for this lane
    const int ncol = ntile * 16 + m;  // global output column for this lane

    const float* Hs[4] = {H0, H1, H2, H3};
    v8f c = {};

#pragma unroll
    for (int s = 0; s < 4; ++s) {
        const float* __restrict__ A  = Hs[s];
        const float* __restrict__ Wb = W + (size_t)s * D * 64;
#pragma unroll
        for (int k = 0; k < D; k += 4) {
            const int kk = k + half * 2;   // this lane's K pair
            v2f a;
            a.x = A[(size_t)arow * D + kk];
            a.y = A[(size_t)arow * D + kk + 1];
            v2f b;
            b.x = Wb[(size_t)kk * 64 + ncol];
            b.y = Wb[(size_t)(kk + 1) * 64 + ncol];
            c = __builtin_amdgcn_wmma_f32_16x16x4_f32(
                    /*neg_a=*/false, a, /*neg_b=*/false, b,
                    /*c_mod=*/(short)0, c, /*reuse_a=*/false, /*reuse_b=*/false);
        }
    }

    const float bv = bias[ncol];
#pragma unroll
    for (int v = 0; v < 8; ++v) {
        const int orow = mtile * 16 + v + half * 8;   // C/D layout: lanes 16-31 = M+8
        float r = c[v] + bv;
        out[(size_t)orow * 64 + ncol] = r > 0.0f ? r : 0.0f;
    }
}

// ---------------- host-side layer driver ----------------

template <int D>
static void run_layer(const float* hin, const float* W, const float* bias,
                      float* buf1, float* buf2, float* buf3,
                      const int* row, const int* col, const float* norm,
                      float* hout, hipStream_t stream) {
    const size_t hopBytes = (size_t)N_NODES * D * sizeof(float);
    const int sblocks = (E_EDGES * (D / 4)) / 256;   // exact

    hipMemsetAsync(buf1, 0, hopBytes, stream);
    scatter_kernel<D><<<sblocks, 256, 0, stream>>>(hin, buf1, row, col, norm);
    hipMemsetAsync(buf2, 0, hopBytes, stream);
    scatter_kernel<D><<<sblocks, 256, 0, stream>>>(buf1, buf2, row, col, norm);
    hipMemsetAsync(buf3, 0, hopBytes, stream);
    scatter_kernel<D><<<sblocks, 256, 0, stream>>>(buf2, buf3, row, col, norm);

    tag_gemm_kernel<D><<<N_NODES / 32, 256, 0, stream>>>(
        hin, buf1, buf2, buf3, W, bias, hout);
}

extern "C" void kernel_launch(void* const* d_in, const int* in_sizes, int n_in,
                              void* d_out, int out_size, void* d_ws, size_t ws_size,
                              hipStream_t stream) {
    const float* x   = (const float*)d_in[0];           // (N, 32)
    const int*   ei  = (const int*)d_in[1];             // (2, E)
    const float* ew  = (const float*)d_in[2];           // (E,)
    const float* W0  = (const float*)d_in[3];           // (4, 32, 64)
    const float* b0  = (const float*)d_in[4];           // (64,)
    const float* Ws  = (const float*)d_in[5];           // (4, 4, 64, 64)
    const float* bs  = (const float*)d_in[6];           // (4, 64)
    float*       out = (float*)d_out;                   // (N, 64)

    const int* row = ei;
    const int* col = ei + E_EDGES;

    // Workspace layout (all offsets 16B-aligned):
    float* ws   = (float*)d_ws;
    float* deg  = ws;                                   // N
    float* norm = deg + N_NODES;                        // E
    float* bufA = norm + E_EDGES;                       // N*64  (layer ping)
    float* bufB = bufA + (size_t)N_NODES * LATENT;      // N*64  (layer pong)
    float* buf1 = bufB + (size_t)N_NODES * LATENT;      // N*64  (hop 1)
    float* buf2 = buf1 + (size_t)N_NODES * LATENT;      // N*64  (hop 2)
    float* buf3 = buf2 + (size_t)N_NODES * LATENT;      // N*64  (hop 3)

    // ---- gcn_norm (symmetric, no self loops) ----
    hipMemsetAsync(deg, 0, N_NODES * sizeof(float), stream);
    deg_kernel<<<E_EDGES / 256, 256, 0, stream>>>(col, ew, deg);
    dinv_kernel<<<(N_NODES + 255) / 256, 256, 0, stream>>>(deg);
    norm_kernel<<<E_EDGES / 256, 256, 0, stream>>>(row, col, ew, deg, norm);

    // ---- 5 TAGConv layers ----
    const size_t WL = 4 * 64 * 64;  // per-layer weight chunk in Ws
    run_layer<IN_DIM>(x,    W0,          b0,       buf1, buf2, buf3, row, col, norm, bufA, stream);
    run_layer<LATENT>(bufA, Ws + 0 * WL, bs + 0,   buf1, buf2, buf3, row, col, norm, bufB, stream);
    run_layer<LATENT>(bufB, Ws + 1 * WL, bs + 64,  buf1, buf2, buf3, row, col, norm, bufA, stream);
    run_layer<LATENT>(bufA, Ws + 2 * WL, bs + 128, buf1, buf2, buf3, row, col, norm, bufB, stream);
    run_layer<LATENT>(bufB, Ws + 3 * WL, bs + 192, buf1, buf2, buf3, row, col, norm, out,  stream);
}